// RNNStackJoulinImproved_39951785788097
// MI455X (gfx1250) — compile-verified
//
#include <hip/hip_runtime.h>
#include <math.h>

// Problem constants (from reference):
#define B_    16      // batch
#define S_    256     // sequence length
#define V_    32000   // vocab
#define E_    128     // embed dim
#define H_    256     // hidden dim
#define NSTK  20      // stack depth
// Workspace requirement: (V*E + V*H + S*B*H + S*B*E + S*B*2 + 2*B*NSTK*E) floats ~= 55.8 MB

typedef float v2f __attribute__((ext_vector_type(2)));
typedef float v8f __attribute__((ext_vector_type(8)));

// D = A(16x4 f32) * B(4x16 f32) + C(16x16 f32), wave32 striped.
__device__ __forceinline__ v8f wmma4(v2f a, v2f b, v8f c) {
  return __builtin_amdgcn_wmma_f32_16x16x4_f32(false, a, false, b, (short)0, c, false, false);
}

__device__ __forceinline__ float gumbel(float u) { return -__logf(-__logf(u)); }

// ---------------------------------------------------------------------------
// Kernel 0: fold W_fc halves (hmod2 trick) and row-normalize emb.
// One wave per vocab row v. grid = V/8, block = 256 (8 waves).
// ---------------------------------------------------------------------------
__global__ void __launch_bounds__(256)
prep_kernel(const float* __restrict__ emb, const float* __restrict__ W_fc,
            float* __restrict__ emb_n, float* __restrict__ wsum) {
  int lane = threadIdx.x & 31;
  int v = blockIdx.x * 8 + (threadIdx.x >> 5);
  // normalized embedding row (clip norm at 1e-8)
  float x[4]; float ss = 0.f;
  #pragma unroll
  for (int j = 0; j < 4; ++j) { x[j] = emb[(size_t)v * E_ + lane * 4 + j]; ss += x[j] * x[j]; }
  #pragma unroll
  for (int m = 1; m < 32; m <<= 1) ss += __shfl_xor(ss, m, 32);
  float rn = 1.0f / fmaxf(sqrtf(ss), 1e-8f);
  #pragma unroll
  for (int j = 0; j < 4; ++j) emb_n[(size_t)v * E_ + lane * 4 + j] = x[j] * rn;
  // W_fc_sum[v][j] = W_fc[v][j] + W_fc[v][j+256]
  #pragma unroll
  for (int k = 0; k < 8; ++k) {
    int j = lane + 32 * k;
    wsum[(size_t)v * H_ + j] = W_fc[(size_t)v * (2 * H_) + j] + W_fc[(size_t)v * (2 * H_) + H_ + j];
  }
}

// ---------------------------------------------------------------------------
// Kernel 1: sequential scan over t. One workgroup, 512 threads = 16 waves.
// Wave w owns batch row b=w for vector phases, and the 16-wide N-tile
// [w*16, w*16+16) of h for the WMMA recurrent GEMM.
// Emits per step: hmod_all[t] (16x256), q_all[t] (16x128), choice_all[t] (16x2).
// Stack double-buffered in global workspace.
// ---------------------------------------------------------------------------
__global__ void __launch_bounds__(512)
scan_kernel(const int* __restrict__ x_ids, const float* __restrict__ emb,
            const float* __restrict__ W_i, const float* __restrict__ W_h,
            const float* __restrict__ W_choose, const float* __restrict__ W_ops,
            const float* __restrict__ gum_ops, const float* __restrict__ gum_choose,
            float* __restrict__ hmod_all, float* __restrict__ q_all,
            float* __restrict__ choice_all, float* __restrict__ stA, float* __restrict__ stB) {
  __shared__ float sh_h[B_ * H_];      // current hidden state (GEMM A operand)
  __shared__ float sh_hmod[B_ * H_];   // 1 - relu(h)
  __shared__ float sh_xi[B_ * E_];     // raw embedding (pushed on stack)
  __shared__ float sh_xin[B_ * E_];    // normalized embedding (GEMM A operand)
  __shared__ float sh_wops[3 * H_];    // folded W_ops
  __shared__ float sh_wch[2 * H_];     // folded W_choose
  __shared__ float sh_opslog[B_ * 3];
  __shared__ float sh_chlog[B_ * 2];
  __shared__ float sh_push[B_], sh_pop[B_], sh_nop[B_];

  const int tid = threadIdx.x, lane = tid & 31, w = tid >> 5;
  const int m = lane & 15, hi = lane >> 4, kb = hi * 2;

  for (int f = tid; f < B_ * H_; f += 512) sh_h[f] = 0.f;
  for (int f = tid; f < 3 * H_; f += 512) {
    int i = f >> 8, j = f & 255;
    sh_wops[f] = W_ops[i * 512 + j] + W_ops[i * 512 + 256 + j];
  }
  if (tid < 2 * H_) {
    int i = tid >> 8, j = tid & 255;
    sh_wch[tid] = W_choose[i * 512 + j] + W_choose[i * 512 + 256 + j];
  }
  for (int f = tid; f < B_ * NSTK * E_; f += 512) stA[f] = 0.f;  // stack0 = zeros
  __threadfence_block();
  __syncthreads();

  for (int t = 0; t < S_; ++t) {
    float* oldS = (t & 1) ? stB : stA;
    float* newS = (t & 1) ? stA : stB;

    // -------- phase 1: gather embedding row, compute raw + normalized (wave w = batch b)
    {
      const int b = w;
      int id = x_ids[b * S_ + t];
      float x[4]; float ss = 0.f;
      #pragma unroll
      for (int j = 0; j < 4; ++j) { x[j] = emb[(size_t)id * E_ + lane * 4 + j]; ss += x[j] * x[j]; }
      #pragma unroll
      for (int msk = 1; msk < 32; msk <<= 1) ss += __shfl_xor(ss, msk, 32);
      float rn = 1.0f / fmaxf(sqrtf(ss), 1e-12f);
      #pragma unroll
      for (int j = 0; j < 4; ++j) {
        sh_xi[b * E_ + lane * 4 + j] = x[j];
        sh_xin[b * E_ + lane * 4 + j] = x[j] * rn;
      }
    }
    __syncthreads();

    // -------- phase 2: hnew = tanh(xin@W_i^T + h@W_h^T), WMMA f32 16x16x4
    v8f acc = {0.f, 0.f, 0.f, 0.f, 0.f, 0.f, 0.f, 0.f};
    {
      const int n = w * 16 + m;  // output column (hidden unit)
      #pragma unroll 8
      for (int i = 0; i < E_ / 4; ++i) {
        v2f a = *(const v2f*)&sh_xin[m * E_ + 4 * i + kb];
        v2f bb = *(const v2f*)&W_i[(size_t)n * E_ + 4 * i + kb];
        acc = wmma4(a, bb, acc);
      }
      #pragma unroll 8
      for (int i = 0; i < H_ / 4; ++i) {
        v2f a = *(const v2f*)&sh_h[m * H_ + 4 * i + kb];
        v2f bb = *(const v2f*)&W_h[(size_t)n * H_ + 4 * i + kb];
        acc = wmma4(a, bb, acc);
      }
    }
    __syncthreads();  // all waves done reading sh_h before anyone rewrites it
    {
      #pragma unroll
      for (int r = 0; r < 8; ++r) {
        int row = r + 8 * hi;        // batch row
        int col = w * 16 + m;        // hidden unit
        float hv = tanhf(acc[r]);
        float hm = 1.0f - fmaxf(hv, 0.f);
        sh_h[row * H_ + col] = hv;
        sh_hmod[row * H_ + col] = hm;
        hmod_all[((size_t)t * B_ + row) * H_ + col] = hm;
      }
    }
    __syncthreads();

    // -------- phase 3: ops/choice logits (folded weights), Gumbel softmax
    if (tid < 48) {
      int b = tid / 3, i = tid % 3;
      float s = 0.f;
      for (int j = 0; j < H_; ++j) s += sh_hmod[b * H_ + j] * sh_wops[i * H_ + j];
      sh_opslog[b * 3 + i] = s;
    } else if (tid < 80) {
      int idx = tid - 48; int b = idx >> 1, i = idx & 1;
      float s = 0.f;
      for (int j = 0; j < H_; ++j) s += sh_hmod[b * H_ + j] * sh_wch[i * H_ + j];
      sh_chlog[b * 2 + i] = s;
    }
    __syncthreads();
    if (tid < B_) {
      int b = tid;
      float l0 = sh_opslog[b * 3 + 0] + gumbel(gum_ops[((size_t)t * B_ + b) * 3 + 0]);
      float l1 = sh_opslog[b * 3 + 1] + gumbel(gum_ops[((size_t)t * B_ + b) * 3 + 1]);
      float l2 = sh_opslog[b * 3 + 2] + gumbel(gum_ops[((size_t)t * B_ + b) * 3 + 2]);
      float mx = fmaxf(l0, fmaxf(l1, l2));
      float e0 = __expf(l0 - mx), e1 = __expf(l1 - mx), e2 = __expf(l2 - mx);
      float inv = 1.0f / (e0 + e1 + e2);
      sh_push[b] = e0 * inv; sh_pop[b] = e1 * inv; sh_nop[b] = e2 * inv;
      float c0 = sh_chlog[b * 2 + 0] + gumbel(gum_choose[((size_t)t * B_ + b) * 2 + 0]);
      float c1 = sh_chlog[b * 2 + 1] + gumbel(gum_choose[((size_t)t * B_ + b) * 2 + 1]);
      float cm = fmaxf(c0, c1);
      float f0 = __expf(c0 - cm), f1 = __expf(c1 - cm);
      float ci = 1.0f / (f0 + f1);
      choice_all[((size_t)t * B_ + b) * 2 + 0] = f0 * ci;
      choice_all[((size_t)t * B_ + b) * 2 + 1] = f1 * ci;
    }
    __syncthreads();

    // -------- phase 4: stack = push*shift_in(xi) + pop*shift_out + nop*stack
    for (int f = tid; f < B_ * NSTK * E_; f += 512) {
      int b = f / (NSTK * E_);
      int rem = f - b * (NSTK * E_);
      int s = rem >> 7;  // slot
      float ps = sh_push[b], pp = sh_pop[b], nn = sh_nop[b];
      float pushed = (s == 0) ? sh_xi[b * E_ + (rem & 127)] : oldS[f - E_];
      float popped = (s == NSTK - 1) ? 0.f : oldS[f + E_];
      newS[f] = ps * pushed + pp * popped + nn * oldS[f];
    }
    __threadfence_block();
    __syncthreads();

    // -------- phase 5: q[b] = sum over top-4 normalized peeks (wave w = batch b)
    {
      const int b = w;
      float qv[4] = {0.f, 0.f, 0.f, 0.f};
      for (int s = 0; s < 4; ++s) {
        const float* p = &newS[b * NSTK * E_ + s * E_];
        float x[4]; float ss = 0.f;
        #pragma unroll
        for (int j = 0; j < 4; ++j) { x[j] = p[lane * 4 + j]; ss += x[j] * x[j]; }
        #pragma unroll
        for (int msk = 1; msk < 32; msk <<= 1) ss += __shfl_xor(ss, msk, 32);
        float rn = 1.0f / fmaxf(sqrtf(ss), 1e-8f);
        #pragma unroll
        for (int j = 0; j < 4; ++j) qv[j] += x[j] * rn;
      }
      #pragma unroll
      for (int j = 0; j < 4; ++j)
        q_all[((size_t)t * B_ + b) * E_ + lane * 4 + j] = qv[j];
    }
    __syncthreads();
  }
}

// ---------------------------------------------------------------------------
// Kernel 2: per-t softmax + blend, parallel over t. grid = S, block = 512 (16 waves).
// Wave w owns 16 vocab columns per 256-wide chunk; 125 chunks cover V.
// Pass 1: WMMA semantic logits + online (max, sumexp). Pass 2: recompute logits,
// WMMA syntactic (q @ emb_n^T), blend with choice, write 524 MB output once.
// ---------------------------------------------------------------------------
__global__ void __launch_bounds__(512)
fuse_kernel(const float* __restrict__ hmod_all, const float* __restrict__ q_all,
            const float* __restrict__ choice_all, const float* __restrict__ wsum,
            const float* __restrict__ emb_n, float* __restrict__ out) {
  const int t = blockIdx.x;
  __shared__ float sh_hm[B_ * H_];
  __shared__ float sh_q[B_ * E_];
  __shared__ float sh_c0[B_], sh_c1[B_];
  __shared__ float sh_rmax[16 * 16], sh_rsum[16 * 16];  // [wave][batch row]
  __shared__ float sh_M[B_], sh_Zi[B_];

  const int tid = threadIdx.x, lane = tid & 31, w = tid >> 5;
  const int m = lane & 15, hi = lane >> 4, kb = hi * 2;

  for (int f = tid; f < B_ * H_; f += 512) sh_hm[f] = hmod_all[(size_t)t * B_ * H_ + f];
  for (int f = tid; f < B_ * E_; f += 512) sh_q[f] = q_all[(size_t)t * B_ * E_ + f];
  if (tid < B_) {
    sh_c0[tid] = choice_all[((size_t)t * B_ + tid) * 2 + 0];
    sh_c1[tid] = choice_all[((size_t)t * B_ + tid) * 2 + 1];
  }
  __syncthreads();

  // ----- pass 1: online softmax statistics over V
  float rmax[8], rsum[8];
  #pragma unroll
  for (int r = 0; r < 8; ++r) { rmax[r] = -1e30f; rsum[r] = 0.f; }

  for (int c = 0; c < V_ / 256; ++c) {
    const int n = c * 256 + w * 16 + m;  // vocab column
    v8f acc = {0.f, 0.f, 0.f, 0.f, 0.f, 0.f, 0.f, 0.f};
    #pragma unroll 8
    for (int i = 0; i < H_ / 4; ++i) {
      v2f a = *(const v2f*)&sh_hm[m * H_ + 4 * i + kb];
      v2f bb = *(const v2f*)&wsum[(size_t)n * H_ + 4 * i + kb];
      acc = wmma4(a, bb, acc);
    }
    #pragma unroll
    for (int r = 0; r < 8; ++r) {
      float tmax = acc[r];
      #pragma unroll
      for (int msk = 1; msk < 16; msk <<= 1) tmax = fmaxf(tmax, __shfl_xor(tmax, msk, 32));
      float nm = fmaxf(rmax[r], tmax);
      float e = __expf(acc[r] - nm);
      #pragma unroll
      for (int msk = 1; msk < 16; msk <<= 1) e += __shfl_xor(e, msk, 32);
      rsum[r] = rsum[r] * __expf(rmax[r] - nm) + e;
      rmax[r] = nm;
    }
  }
  if ((lane & 15) == 0) {
    #pragma unroll
    for (int r = 0; r < 8; ++r) {
      sh_rmax[w * 16 + r + 8 * hi] = rmax[r];
      sh_rsum[w * 16 + r + 8 * hi] = rsum[r];
    }
  }
  __syncthreads();
  if (tid < B_) {
    float M = -1e30f;
    for (int w2 = 0; w2 < 16; ++w2) M = fmaxf(M, sh_rmax[w2 * 16 + tid]);
    float Z = 0.f;
    for (int w2 = 0; w2 < 16; ++w2) Z += sh_rsum[w2 * 16 + tid] * __expf(sh_rmax[w2 * 16 + tid] - M);
    sh_M[tid] = M;
    sh_Zi[tid] = 1.0f / Z;
  }
  __syncthreads();

  // ----- pass 2: recompute logits (cheaper than spilling 2MB/t), add syntactic, write
  for (int c = 0; c < V_ / 256; ++c) {
    const int n = c * 256 + w * 16 + m;
    v8f acc = {0.f, 0.f, 0.f, 0.f, 0.f, 0.f, 0.f, 0.f};
    v8f sacc = {0.f, 0.f, 0.f, 0.f, 0.f, 0.f, 0.f, 0.f};
    #pragma unroll 8
    for (int i = 0; i < H_ / 4; ++i) {
      v2f a = *(const v2f*)&sh_hm[m * H_ + 4 * i + kb];
      v2f bb = *(const v2f*)&wsum[(size_t)n * H_ + 4 * i + kb];
      acc = wmma4(a, bb, acc);
    }
    #pragma unroll 8
    for (int i = 0; i < E_ / 4; ++i) {
      v2f a = *(const v2f*)&sh_q[m * E_ + 4 * i + kb];
      v2f bb = *(const v2f*)&emb_n[(size_t)n * E_ + 4 * i + kb];
      sacc = wmma4(a, bb, sacc);
    }
    #pragma unroll
    for (int r = 0; r < 8; ++r) {
      int row = r + 8 * hi;  // batch row
      float sem = __expf(acc[r] - sh_M[row]) * sh_Zi[row];
      float val = sh_c0[row] * sem + sh_c1[row] * sacc[r];
      out[((size_t)row * S_ + t) * V_ + n] = val;  // out is (B, S, V)
    }
  }
}

// ---------------------------------------------------------------------------
extern "C" void kernel_launch(void* const* d_in, const int* in_sizes, int n_in,
                              void* d_out, int out_size, void* d_ws, size_t ws_size,
                              hipStream_t stream) {
  const int* x_ids        = (const int*)d_in[0];
  const float* emb        = (const float*)d_in[1];
  const float* W_i        = (const float*)d_in[2];
  const float* W_h        = (const float*)d_in[3];
  const float* W_fc       = (const float*)d_in[4];
  const float* W_choose   = (const float*)d_in[5];
  const float* W_ops      = (const float*)d_in[6];
  const float* gum_ops    = (const float*)d_in[7];
  const float* gum_choose = (const float*)d_in[8];
  float* out = (float*)d_out;
  float* ws = (float*)d_ws;

  float* emb_n    = ws;                                 // V*E
  float* wsum     = emb_n + (size_t)V_ * E_;            // V*H
  float* hmod_all = wsum + (size_t)V_ * H_;             // S*B*H
  float* q_all    = hmod_all + (size_t)S_ * B_ * H_;    // S*B*E
  float* ch_all   = q_all + (size_t)S_ * B_ * E_;       // S*B*2
  float* stA      = ch_all + (size_t)S_ * B_ * 2;       // B*NSTK*E
  float* stB      = stA + (size_t)B_ * NSTK * E_;       // B*NSTK*E

  prep_kernel<<<V_ / 8, 256, 0, stream>>>(emb, W_fc, emb_n, wsum);
  scan_kernel<<<1, 512, 0, stream>>>(x_ids, emb, W_i, W_h, W_choose, W_ops,
                                     gum_ops, gum_choose, hmod_all, q_all, ch_all, stA, stB);
  fuse_kernel<<<S_, 512, 0, stream>>>(hmod_all, q_all, ch_all, wsum, emb_n, out);
}